// Model_1494648619082
// MI455X (gfx1250) — compile-verified
//
#include <hip/hip_runtime.h>

// Problem sizes (match reference)
#define BROWS   65536
#define DDIM    768
#define DCDIM   192
#define KCB     100
#define KPAD    112      // 7 tiles of 16
#define ROWS    32       // rows per block
#define THREADS 256      // 8 wave32
#define KCHUNK  16       // K-chunk per pipeline stage
#define NCHUNK  (DDIM / KCHUNK)   // 48
#define ASTR    20       // LDS stride (floats): conflict-free, 8B aligned
#define LN_EPS  1e-5f

typedef __attribute__((ext_vector_type(2))) float v2f;
typedef __attribute__((ext_vector_type(4))) float v4f;
typedef __attribute__((ext_vector_type(8))) float v8f;

// Async global -> LDS copy (8 bytes per lane), tracked by ASYNCcnt.
__device__ __forceinline__ void async_copy_b64(unsigned ldsOff, unsigned long long gaddr) {
  asm volatile("global_load_async_to_lds_b64 %0, %1, off"
               :: "v"(ldsOff), "v"(gaddr) : "memory");
}
__device__ __forceinline__ void wait_async_le7() {
  asm volatile("s_wait_asynccnt 0x7" ::: "memory");
}
__device__ __forceinline__ void wait_async_0() {
  asm volatile("s_wait_asynccnt 0x0" ::: "memory");
}

// ---------------- Precompute kernels (tiny, run once per launch) ----------------

// k_mat[j][e] = sum_c codebook[j,c] * Wk[e,c]          (100 x 192)
__global__ __launch_bounds__(DCDIM) void precompute_kmat(
    const float* __restrict__ codebook, const float* __restrict__ Wk,
    float* __restrict__ k_mat) {
  const int j = blockIdx.x, e = threadIdx.x;
  float s = 0.f;
  #pragma unroll 4
  for (int c = 0; c < DCDIM; ++c)
    s += codebook[j * DCDIM + c] * Wk[e * DCDIM + c];
  k_mat[j * DCDIM + e] = s;
}

// Mt[j][d] = sum_e Wq[e,d] * k_mat[j,e]   (112 x 192, rows >= 100 zeroed)
__global__ __launch_bounds__(DCDIM) void precompute_mt(
    const float* __restrict__ Wq, const float* __restrict__ k_mat,
    float* __restrict__ Mt) {
  const int j = blockIdx.x, d = threadIdx.x;
  float s = 0.f;
  if (j < KCB) {
    #pragma unroll 4
    for (int e = 0; e < DCDIM; ++e)
      s += Wq[e * DCDIM + d] * k_mat[j * DCDIM + e];
  }
  Mt[j * DCDIM + d] = s;
}

// code_high[j][d] = b_high[d] + sum_c codebook[j,c] * W_high[d,c]   (100 x 768)
__global__ __launch_bounds__(THREADS) void precompute_codehigh(
    const float* __restrict__ codebook, const float* __restrict__ W_high,
    const float* __restrict__ b_high, float* __restrict__ code_high) {
  const int j = blockIdx.x;
  for (int d = threadIdx.x; d < DDIM; d += THREADS) {
    float s = b_high[d];
    #pragma unroll 4
    for (int c = 0; c < DCDIM; ++c)
      s += codebook[j * DCDIM + c] * W_high[d * DCDIM + c];
    code_high[j * DDIM + d] = s;
  }
}

// ---------------- Main fused kernel ----------------
// Per block (32 rows):
//  1) T = hidden[32,768] @ W_low^T : WMMA f32 16x16x4 with double-buffered
//     GLOBAL_LOAD_ASYNC_TO_LDS_B64 staging (ASYNCcnt pipeline)
//  2) LayerNorm rows of T (8 threads/row LDS reduction)
//  3) S = pat @ Mt^T, idx = argmax_j (WMMA, Mt frags from L2)
//  4) out[row] = code_high[idx] (float4 gather)

__global__ __launch_bounds__(THREADS) void fused_vq_forward(
    const float* __restrict__ hidden, const float* __restrict__ W_low,
    const float* __restrict__ b_low, const float* __restrict__ ln_g,
    const float* __restrict__ ln_b, const float* __restrict__ Mt,
    const float* __restrict__ code_high, float* __restrict__ out) {
  // LDS: 65,408 B total (<= 64 KB). Stride 20 floats -> conflict-free b64 frags.
  __shared__ float As[2][ROWS * ASTR];    // hidden k-chunks (double buffered)
  __shared__ float BsU[2][DCDIM * ASTR];  // W_low k-chunks; later reused as Ss[32][114]
  __shared__ float Ts[ROWS * 194];        // low-proj output / pat
  __shared__ float blS[DCDIM], lgS[DCDIM], lbS[DCDIM];
  __shared__ float red1[ROWS * 8], red2[ROWS * 8];
  __shared__ float rowMu[ROWS], rowRs[ROWS];
  __shared__ int   idxs[ROWS];

  const int tid  = threadIdx.x;
  const int lane = tid & 31;
  const int wave = tid >> 5;
  const int l15  = lane & 15;
  const int koff = (lane < 16) ? 0 : 2;     // A/B frag K-offset per lane half
  const int rowBase = blockIdx.x * ROWS;

  if (tid < DCDIM) { blS[tid] = b_low[tid]; lgS[tid] = ln_g[tid]; lbS[tid] = ln_b[tid]; }

  // Wave tiling: strip = 16-row half, grp = 3 column tiles (of 12)
  const int strip = wave & 1;
  const int grp   = wave >> 1;              // 0..3
  const int r0    = strip * 16;

  // ---- Async copy plan: 1792 b64 transfers/chunk = 7 per thread.
  //      t==0: the 256 hidden b64s; t in 1..6: the 1536 W_low b64s.
  unsigned long long gBase[7];
  unsigned ldsBase[7];
  unsigned bufStride[7];
  {
    int r = tid >> 3, c = tid & 7;
    gBase[0]     = (unsigned long long)(const void*)&hidden[(size_t)(rowBase + r) * DDIM + 2 * c];
    ldsBase[0]   = (unsigned)(size_t)(void*)&As[0][r * ASTR + 2 * c];
    bufStride[0] = ROWS * ASTR * 4;
    #pragma unroll
    for (int t = 1; t < 7; ++t) {
      int j = tid + 256 * (t - 1);          // 0..1535
      int rr = j >> 3, cc = j & 7;          // rr: 0..191
      gBase[t]     = (unsigned long long)(const void*)&W_low[(size_t)rr * DDIM + 2 * cc];
      ldsBase[t]   = (unsigned)(size_t)(void*)&BsU[0][rr * ASTR + 2 * cc];
      bufStride[t] = DCDIM * ASTR * 4;
    }
  }
  #define ISSUE_CHUNK(ch, buf)                                              \
    {                                                                       \
      _Pragma("unroll")                                                     \
      for (int t = 0; t < 7; ++t)                                           \
        async_copy_b64(ldsBase[t] + (unsigned)(buf) * bufStride[t],         \
                       gBase[t] + (unsigned long long)(ch) * (KCHUNK * 4)); \
    }

  v8f acc0 = {}, acc1 = {}, acc2 = {};

  // ---- Phase 1: T = hidden @ W_low^T, double-buffered async pipeline ----
  ISSUE_CHUNK(0, 0);
  ISSUE_CHUNK(1, 1);
  for (int ch = 0; ch < NCHUNK; ++ch) {
    const int buf = ch & 1;
    if (ch + 1 < NCHUNK) wait_async_le7();  // chunk `ch` landed; `ch+1` may fly
    else                 wait_async_0();
    __syncthreads();                        // LDS visible to all waves
    const int c0 = grp * 48;
    #pragma unroll
    for (int kk = 0; kk < KCHUNK; kk += 4) {
      v2f a  = *(const v2f*)&As[buf][(r0 + l15) * ASTR + kk + koff];
      v2f b0 = *(const v2f*)&BsU[buf][(c0      + l15) * ASTR + kk + koff];
      acc0 = __builtin_amdgcn_wmma_f32_16x16x4_f32(false, a, false, b0,
                                                   (short)0, acc0, false, false);
      v2f b1 = *(const v2f*)&BsU[buf][(c0 + 16 + l15) * ASTR + kk + koff];
      acc1 = __builtin_amdgcn_wmma_f32_16x16x4_f32(false, a, false, b1,
                                                   (short)0, acc1, false, false);
      v2f b2 = *(const v2f*)&BsU[buf][(c0 + 32 + l15) * ASTR + kk + koff];
      acc2 = __builtin_amdgcn_wmma_f32_16x16x4_f32(false, a, false, b2,
                                                   (short)0, acc2, false, false);
    }
    __syncthreads();                        // all waves done reading `buf`
    if (ch + 2 < NCHUNK) ISSUE_CHUNK(ch + 2, buf);
  }
  // Spill C tiles to Ts. C layout: VGPR i -> M = i (lanes 0-15) / 8+i (16-31)
  {
    const int mbase = r0 + ((lane < 16) ? 0 : 8);
    const int cc = grp * 48 + l15;
    #pragma unroll
    for (int i = 0; i < 8; ++i) {
      Ts[(mbase + i) * 194 + cc]      = acc0[i];
      Ts[(mbase + i) * 194 + cc + 16] = acc1[i];
      Ts[(mbase + i) * 194 + cc + 32] = acc2[i];
    }
  }
  __syncthreads();

  // ---- Phase 2: LayerNorm (8 threads per row) ----
  {
    const int r = tid >> 3, sub = tid & 7;
    float s1 = 0.f, s2 = 0.f;
    #pragma unroll 4
    for (int q = 0; q < 24; ++q) {
      int d = sub * 24 + q;
      float x = Ts[r * 194 + d] + blS[d];
      s1 += x; s2 += x * x;
    }
    red1[r * 8 + sub] = s1; red2[r * 8 + sub] = s2;
    __syncthreads();
    if (sub == 0) {
      float a = 0.f, b = 0.f;
      #pragma unroll
      for (int q = 0; q < 8; ++q) { a += red1[r * 8 + q]; b += red2[r * 8 + q]; }
      float mu  = a * (1.f / DCDIM);
      float var = b * (1.f / DCDIM) - mu * mu;
      rowMu[r] = mu;
      rowRs[r] = rsqrtf(var + LN_EPS);
    }
    __syncthreads();
    const float mu = rowMu[r], rs = rowRs[r];
    #pragma unroll 4
    for (int q = 0; q < 24; ++q) {
      int d = sub * 24 + q;
      Ts[r * 194 + d] = (Ts[r * 194 + d] + blS[d] - mu) * rs * lgS[d] + lbS[d];
    }
  }
  __syncthreads();

  // ---- Phase 3: S = pat @ Mt^T (7 col tiles over 4 wave-groups: 2/2/2/1) ----
  v8f sA = {}, sB = {};
  const int ntile = (grp < 3) ? 2 : 1;
  const int jt0   = grp * 2;
  for (int d0 = 0; d0 < DCDIM; d0 += 4) {
    v2f a = *(const v2f*)&Ts[(r0 + l15) * 194 + d0 + koff];
    {
      int j = jt0 * 16 + l15;
      v2f b = *(const v2f*)&Mt[(size_t)j * DCDIM + d0 + koff];
      sA = __builtin_amdgcn_wmma_f32_16x16x4_f32(false, a, false, b,
                                                 (short)0, sA, false, false);
    }
    if (ntile == 2) {
      int j = (jt0 + 1) * 16 + l15;
      v2f b = *(const v2f*)&Mt[(size_t)j * DCDIM + d0 + koff];
      sB = __builtin_amdgcn_wmma_f32_16x16x4_f32(false, a, false, b,
                                                 (short)0, sB, false, false);
    }
  }
  // Scores into BsU reused as Ss[32][114]
  float* Ss = &BsU[0][0];
  {
    const int mbase = r0 + ((lane < 16) ? 0 : 8);
    #pragma unroll
    for (int i = 0; i < 8; ++i) {
      Ss[(mbase + i) * 114 + jt0 * 16 + l15] = sA[i];
      if (ntile == 2)
        Ss[(mbase + i) * 114 + (jt0 + 1) * 16 + l15] = sB[i];
    }
  }
  __syncthreads();

  // ---- Phase 4: per-row argmax (argmin of attn_w == argmax of q.k) ----
  if (tid < ROWS) {
    float best = Ss[tid * 114 + 0];
    int bi = 0;
    for (int j = 1; j < KCB; ++j) {
      float v = Ss[tid * 114 + j];
      if (v > best) { best = v; bi = j; }
    }
    idxs[tid] = bi;
  }
  __syncthreads();

  // ---- Phase 5: gather out[row] = code_high[idx] (float4) ----
  const v4f* __restrict__ ch4 = (const v4f*)code_high;
  v4f* __restrict__ out4 = (v4f*)out;
  for (int i = tid; i < ROWS * (DDIM / 4); i += THREADS) {
    int r = i / (DDIM / 4), c = i % (DDIM / 4);
    out4[(size_t)(rowBase + r) * (DDIM / 4) + c] =
        ch4[(size_t)idxs[r] * (DDIM / 4) + c];
  }
}

// ---------------- Host launcher ----------------
extern "C" void kernel_launch(void* const* d_in, const int* in_sizes, int n_in,
                              void* d_out, int out_size, void* d_ws, size_t ws_size,
                              hipStream_t stream) {
  const float* hidden   = (const float*)d_in[0];
  const float* W_low    = (const float*)d_in[1];
  const float* b_low    = (const float*)d_in[2];
  const float* ln_g     = (const float*)d_in[3];
  const float* ln_b     = (const float*)d_in[4];
  const float* Wq       = (const float*)d_in[5];
  const float* Wk       = (const float*)d_in[6];
  // d_in[7] = Wv, d_in[8] = Wo : drop out of the forward value (STE)
  const float* codebook = (const float*)d_in[9];
  const float* W_high   = (const float*)d_in[10];
  const float* b_high   = (const float*)d_in[11];
  float* out = (float*)d_out;

  // Workspace layout (floats): k_mat[100*192] | Mt[112*192] | code_high[100*768]
  float* ws        = (float*)d_ws;
  float* k_mat     = ws;
  float* Mt        = ws + KCB * DCDIM;
  float* code_high = Mt + KPAD * DCDIM;

  precompute_kmat<<<KCB, DCDIM, 0, stream>>>(codebook, Wk, k_mat);
  precompute_mt<<<KPAD, DCDIM, 0, stream>>>(Wq, k_mat, Mt);
  precompute_codehigh<<<KCB, THREADS, 0, stream>>>(codebook, W_high, b_high, code_high);

  fused_vq_forward<<<BROWS / ROWS, THREADS, 0, stream>>>(
      hidden, W_low, b_low, ln_g, ln_b, Mt, code_high, out);
}